// IoU3DLoss_15333033247305
// MI455X (gfx1250) — compile-verified
//
#include <hip/hip_runtime.h>

typedef __attribute__((ext_vector_type(2))) float v2f;
typedef __attribute__((ext_vector_type(8))) float v8f;

#define BLOCK 1024
#define WAVES (BLOCK / 32)

__global__ __launch_bounds__(BLOCK) void iou3d_diag_loss_kernel(
    const float* __restrict__ pred, const float* __restrict__ target,
    float* __restrict__ out, int m) {
  __shared__ float wave_part[WAVES];

  const int tid = threadIdx.x;
  float acc = 0.0f;

  // Grid-stride within the single block: coalesced-ish row reads (28B/row,
  // total 229KB -> bandwidth-irrelevant; this kernel is latency bound).
  for (int i = tid; i < m; i += BLOCK) {
    const float* p = pred + i * 7;
    const float* t = target + i * 7;
    float px = p[0], py = p[1], pz = p[2], pw = p[3], pl = p[4], ph = p[5], pr = p[6];
    float tx = t[0], ty = t[1], tz = t[2], tw = t[3], tl = t[4], th = t[5], tr = t[6];

    // Rotated-rect AABB half extents (exact match of corner min/max):
    // x' = x*c + y*s, y' = -x*s + y*c over corners (+-w/2, +-l/2)
    float pc = __cosf(pr), psn = __sinf(pr);
    float tc = __cosf(tr), tsn = __sinf(tr);
    float phx = 0.5f * (fabsf(pw * pc) + fabsf(pl * psn));
    float phy = 0.5f * (fabsf(pw * psn) + fabsf(pl * pc));
    float thx = 0.5f * (fabsf(tw * tc) + fabsf(tl * tsn));
    float thy = 0.5f * (fabsf(tw * tsn) + fabsf(tl * tc));

    // BEV overlap (OFFSET = 1e-6 added before clamp, per reference)
    float ltx = fmaxf(px - phx, tx - thx);
    float lty = fmaxf(py - phy, ty - thy);
    float rbx = fminf(px + phx, tx + thx);
    float rby = fminf(py + phy, ty + thy);
    float wx = fmaxf(rbx - ltx + 1e-6f, 0.0f);
    float wy = fmaxf(rby - lty + 1e-6f, 0.0f);
    float ov_bev = wx * wy;

    // Height overlap: box spans [z-h, z]
    float ov_h = fmaxf(fminf(pz, tz) - fmaxf(pz - ph, tz - th), 0.0f);
    float ov3 = ov_bev * ov_h;

    float pv = pw * pl * ph;
    float tv = tw * tl * th;
    float denom = fmaxf(pv + tv - ov3, 1e-6f);
    float iou = fminf(fmaxf(ov3 / denom, 1e-6f), 1.0f);
    acc += -__logf(iou);
  }

  // ---- Wave reduction via V_WMMA_F32_16X16X4_F32 ----
  // A (16x4 f32, 2 VGPRs): lanes 0-15 hold K=0 (vgpr0) / K=1 (vgpr1),
  // lanes 16-31 hold K=2 / K=3.  Put this lane's partial in vgpr0, zero vgpr1:
  // A[m][k] covers all 32 lane partials. B = ones(4x16) => D[m][n] = rowsum(A,m).
  v2f a;
  a.x = acc;
  a.y = 0.0f;
  v2f b;
  b.x = 1.0f;
  b.y = 1.0f;
  v8f c = {};
  v8f d = __builtin_amdgcn_wmma_f32_16x16x4_f32(
      /*neg_a=*/false, a, /*neg_b=*/false, b,
      /*c_mod=*/(short)0, c, /*reuse_a=*/false, /*reuse_b=*/false);

  // Per lane: d[j] = rowsum for M = j (lanes 0-15) or M = 8+j (lanes 16-31).
  float tsum = 0.0f;
#pragma unroll
  for (int j = 0; j < 8; ++j) tsum += d[j];
  // lanes 0-15 hold sum of rows 0-7, lanes 16-31 hold sum of rows 8-15
  tsum += __shfl_xor(tsum, 16, 32);  // full 32-lane wave sum, all lanes

  const int lane = tid & 31;
  const int wave = tid >> 5;
  if (lane == 0) wave_part[wave] = tsum;
  __syncthreads();

  if (wave == 0) {
    float v = wave_part[lane];  // WAVES == 32
    v += __shfl_xor(v, 16, 32);
    v += __shfl_xor(v, 8, 32);
    v += __shfl_xor(v, 4, 32);
    v += __shfl_xor(v, 2, 32);
    v += __shfl_xor(v, 1, 32);
    if (lane == 0) out[0] = v * (1.0f / (float)m);
  }
}

extern "C" void kernel_launch(void* const* d_in, const int* in_sizes, int n_in,
                              void* d_out, int out_size, void* d_ws, size_t ws_size,
                              hipStream_t stream) {
  const float* pred = (const float*)d_in[0];
  const float* target = (const float*)d_in[1];
  float* out = (float*)d_out;
  const int m = in_sizes[0] / 7;  // 4096 in the reference setup
  iou3d_diag_loss_kernel<<<1, BLOCK, 0, stream>>>(pred, target, out, m);
}